// GraphSAGELayer_20916490731750
// MI455X (gfx1250) — compile-verified
//
#include <hip/hip_runtime.h>

// GraphSAGE layer on MI455X (gfx1250, wave32).
// B=4, N=512, C=128, OUT=128.
// Kernel 1: masked max-pool aggregation + ReLU  -> neigh (workspace)
// Kernel 2: [feat | neigh] @ W + b via V_WMMA_F32_16X16X4_F32

#define BB   4
#define NN   512
#define CC   128
#define OUTC 128
#define ROWS (BB * NN)     // 2048

typedef __attribute__((ext_vector_type(2))) float v2f;
typedef __attribute__((ext_vector_type(8))) float v8f;

// ---------------------------------------------------------------------------
// Kernel 1: per-row masked max over neighbors.
// One wave (32 lanes) per output row; each lane owns 4 channels (float4).
// Adjacency row staged in LDS; uniform per-wave branch skips ~90% of loads
// (10% dense adjacency).
// ---------------------------------------------------------------------------
__global__ __launch_bounds__(128) void sage_agg_kernel(
    const float* __restrict__ adj,    // [B, N, N]
    const float* __restrict__ feat,   // [B, N, C]
    float* __restrict__ neigh)        // [B*N, C] (workspace)
{
    __shared__ float sadj[4][NN];     // 4 waves * 512 floats = 8 KB

    const int wave = threadIdx.x >> 5;
    const int lane = threadIdx.x & 31;
    const int row  = blockIdx.x * 4 + wave;      // 0 .. 2047
    const int b    = row >> 9;                   // row / N

    // Stage this wave's adjacency row into LDS (128 float4 / 32 lanes).
    {
        const float4* src4 = (const float4*)(adj + (size_t)row * NN);
        float4* dst4 = (float4*)sadj[wave];
#pragma unroll
        for (int i = 0; i < 4; ++i)
            dst4[lane + 32 * i] = src4[lane + 32 * i];
    }
    __syncthreads();

    const float4* fb = (const float4*)(feat + (size_t)b * NN * CC);

    float v0 = -__builtin_inff();
    float v1 = -__builtin_inff();
    float v2 = -__builtin_inff();
    float v3 = -__builtin_inff();

    for (int m = 0; m < NN; ++m) {
        const float a = sadj[wave][m];           // LDS broadcast: uniform in wave
        if (a > 0.0f) {                          // uniform -> s_cbranch_execz skip
            const float4 f = fb[(size_t)m * (CC / 4) + lane];
            v0 = fmaxf(v0, f.x);
            v1 = fmaxf(v1, f.y);
            v2 = fmaxf(v2, f.z);
            v3 = fmaxf(v3, f.w);
        }
    }

    // ReLU; also maps the no-neighbor (-inf) case to 0, matching the
    // reference's finfo.min / isfinite / relu composition.
    float4 r;
    r.x = fmaxf(v0, 0.0f);
    r.y = fmaxf(v1, 0.0f);
    r.z = fmaxf(v2, 0.0f);
    r.w = fmaxf(v3, 0.0f);
    ((float4*)(neigh + (size_t)row * CC))[lane] = r;
}

// ---------------------------------------------------------------------------
// Kernel 2: out[2048,128] = feat @ W[0:128,:] + neigh @ W[128:256,:] + bias
// One wave per 16x16 output tile; K stepped by 4 with f32 WMMA.
//
// Fragment layouts (CDNA5 ISA 7.12.2, wave32):
//   A 16x4 :  lane group g = lane>>4, M = lane&15, VGPR r -> K = r + 2g
//   B 4x16 :  lane group g,          N = lane&15, VGPR r -> K = r + 2g
//   C/D    :  N = lane&15, VGPR r -> M = r + 8g
// ---------------------------------------------------------------------------
__global__ __launch_bounds__(128) void sage_gemm_kernel(
    const float* __restrict__ feat,   // [2048, 128]
    const float* __restrict__ neigh,  // [2048, 128]
    const float* __restrict__ W,      // [256, 128]
    const float* __restrict__ bias,   // [128]
    float* __restrict__ out)          // [2048, 128]
{
    const int wave = threadIdx.x >> 5;
    const int lane = threadIdx.x & 31;
    const int tile = blockIdx.x * 4 + wave;      // 0 .. 1023
    const int mt   = tile >> 3;                  // 0 .. 127  (row tiles)
    const int nt   = tile & 7;                   // 0 .. 7    (col tiles)
    const int row0 = mt * 16;
    const int col0 = nt * 16;

    const int g  = lane >> 4;                    // lane group (0/1)
    const int lr = lane & 15;
    const int col = col0 + lr;

    // Accumulator initialized with bias (bias depends only on column).
    const float bv = bias[col];
    v8f acc;
#pragma unroll
    for (int i = 0; i < 8; ++i) acc[i] = bv;

    const float* Af = feat  + (size_t)(row0 + lr) * CC;   // A row for this lane
    const float* An = neigh + (size_t)(row0 + lr) * CC;

    // --- first half of K: features x W[0:128, :] ---
#pragma unroll 4
    for (int k0 = 0; k0 < CC; k0 += 4) {
        const int ka = k0 + 2 * g;
        const v2f a = *(const v2f*)(Af + ka);             // A[M][ka], A[M][ka+1]
        v2f b;
        b[0] = W[(size_t)(ka)     * OUTC + col];          // B[ka][N]
        b[1] = W[(size_t)(ka + 1) * OUTC + col];          // B[ka+1][N]
        acc = __builtin_amdgcn_wmma_f32_16x16x4_f32(
                  /*neg_a=*/false, a, /*neg_b=*/false, b,
                  /*c_mod=*/(short)0, acc,
                  /*reuse_a=*/false, /*reuse_b=*/false);
    }

    // --- second half of K: neigh x W[128:256, :] ---
#pragma unroll 4
    for (int k0 = 0; k0 < CC; k0 += 4) {
        const int ka = k0 + 2 * g;
        const v2f a = *(const v2f*)(An + ka);
        v2f b;
        b[0] = W[(size_t)(CC + ka)     * OUTC + col];
        b[1] = W[(size_t)(CC + ka + 1) * OUTC + col];
        acc = __builtin_amdgcn_wmma_f32_16x16x4_f32(
                  false, a, false, b, (short)0, acc, false, false);
    }

    // Store D: M = r + 8*g, N = col.
#pragma unroll
    for (int r = 0; r < 8; ++r) {
        const int row = row0 + r + 8 * g;
        out[(size_t)row * OUTC + col] = acc[r];
    }
}

// ---------------------------------------------------------------------------
extern "C" void kernel_launch(void* const* d_in, const int* in_sizes, int n_in,
                              void* d_out, int out_size, void* d_ws, size_t ws_size,
                              hipStream_t stream) {
    const float* adj  = (const float*)d_in[0];   // [4, 512, 512]
    const float* feat = (const float*)d_in[1];   // [4, 512, 128]
    const float* W    = (const float*)d_in[2];   // [256, 128]
    const float* bias = (const float*)d_in[3];   // [128]
    float* out   = (float*)d_out;                // [4, 512, 128]
    float* neigh = (float*)d_ws;                 // 2048*128 f32 = 1 MB scratch

    // Kernel 1: 2048 rows, 4 waves/block -> 512 blocks of 128 threads.
    sage_agg_kernel<<<512, 128, 0, stream>>>(adj, feat, neigh);

    // Kernel 2: 1024 tiles, 4 waves/block -> 256 blocks of 128 threads.
    sage_gemm_kernel<<<256, 128, 0, stream>>>(feat, neigh, W, bias, out);
}